// MonotoneLayer_14551349199298
// MI455X (gfx1250) — compile-verified
//
#include <hip/hip_runtime.h>

#define NQ   20
#define TPB  256

typedef float v2f __attribute__((ext_vector_type(2)));
typedef float v8f __attribute__((ext_vector_type(8)));

// Async-copy one row (8B of x) into this lane's LDS slot (GVS addressing).
// Offset clamped so we never read past x; clamped lanes are never stored.
__device__ __forceinline__ void async_stage_row(const void* x, int row_idx,
                                                unsigned max_voff,
                                                unsigned lds_off) {
    unsigned voff = (unsigned)row_idx * 8u;
    if (voff > max_voff) voff = max_voff;
    asm volatile("global_load_async_to_lds_b64 %0, %1, %2"
                 :: "v"(lds_off), "v"(voff), "s"(x)
                 : "memory");
}

__global__ __launch_bounds__(TPB) void fused_mono_kernel(
    const float2* __restrict__ x,
    const float* __restrict__ m0_wd,  const float* __restrict__ m0_ws1,
    const float* __restrict__ m0_bs1, const float* __restrict__ m0_wlast,
    const float* __restrict__ m0_blast,
    const float* __restrict__ m1_wd,  const float* __restrict__ m1_ws0,
    const float* __restrict__ m1_bs0, const float* __restrict__ m1_ws1,
    const float* __restrict__ m1_bs1, const float* __restrict__ m1_wlast,
    const float* __restrict__ m1_blast,
    const float* __restrict__ g0_b,
    const float* __restrict__ g1_ws0, const float* __restrict__ g1_bs0,
    const float* __restrict__ g1_ws1, const float* __restrict__ g1_bs1,
    const float* __restrict__ g1_wlast, const float* __restrict__ g1_blast,
    float2* __restrict__ out, int n_rows)
{
    const int tid  = threadIdx.x;
    const int lane = tid & 31;
    const int m    = lane & 15;          // column / row-within-group index
    const bool lo  = lane < 16;          // lane half: supplies k{0,1} vs k{2,3}
    const int ntiles  = (n_rows + TPB - 1) / TPB;
    const int tstride = (int)gridDim.x;

    __shared__ v2f sbuf[2][TPB];
    const unsigned lds0 = (unsigned)(unsigned long long)&sbuf[0][tid];
    const unsigned lds1 = (unsigned)(unsigned long long)&sbuf[1][tid];
    const unsigned max_voff = (unsigned)(n_rows * 8 - 8);

    // ---- hoist weights (uniform addresses -> SMEM/SGPRs) ----
    float w0d[4], w0s1[16], w0b1[4], w0l[4];
    float w1d[4], w1s0[4], w1b0[4], w1b1[4], w1l[4];
    float gw0[8], gb0[4], gw1[16], gb1[4], gwl[4];
#pragma unroll
    for (int i = 0; i < 4; ++i) {
        w0d[i] = m0_wd[i];  w0b1[i] = m0_bs1[i];  w0l[i] = m0_wlast[i];
        w1d[i] = m1_wd[i];  w1s0[i] = m1_ws0[i];  w1b0[i] = m1_bs0[i];
        w1b1[i] = m1_bs1[i]; w1l[i] = m1_wlast[i];
        gb0[i] = g1_bs0[i]; gb1[i] = g1_bs1[i];   gwl[i] = g1_wlast[i];
    }
#pragma unroll
    for (int i = 0; i < 16; ++i) { w0s1[i] = m0_ws1[i]; gw1[i] = g1_ws1[i]; }
#pragma unroll
    for (int i = 0; i < 8; ++i)  { gw0[i] = g1_ws0[i]; }
    const float w0bl = m0_blast[0];
    const float w1bl = m1_blast[0];
    const float g0bv = g0_b[0];
    const float gblv = g1_blast[0];

    // ---- constant WMMA operands (built once) ----
    // A = W1^T, 16x4, rows >=4 zero. A-layout: lane L(<16): M=L, K={0,1} in
    // vgpr{0,1}; lane L(>=16): M=L-16, K={2,3}.  A[m,k] = W1[k,m] = m1_ws1[4k+m].
    const int kb = lo ? 0 : 2;
    v2f Av; Av.x = 0.0f; Av.y = 0.0f;
    if (m < 4) {
        Av.x = m1_ws1[(kb + 0) * 4 + m];
        Av.y = m1_ws1[(kb + 1) * 4 + m];
    }
    // C = bias: D/C layout vgpr r: lanes0-15 = (M=r, N=lane); lanes16-31 =
    // (M=r+8,...) which multiplies A's zero rows -> keep 0 there.
    v8f Cb = {};
    Cb[0] = lo ? w1b1[0] : 0.0f;
    Cb[1] = lo ? w1b1[1] : 0.0f;
    Cb[2] = lo ? w1b1[2] : 0.0f;
    Cb[3] = lo ? w1b1[3] : 0.0f;
    // Per-lane-half mono1 first-layer weights (k = kb, kb+1).
    const float wdA = lo ? w1d[0]  : w1d[2];
    const float wdB = lo ? w1d[1]  : w1d[3];
    const float s0A = lo ? w1s0[0] : w1s0[2];
    const float s0B = lo ? w1s0[1] : w1s0[3];
    const float b0A = lo ? w1b0[0] : w1b0[2];
    const float b0B = lo ? w1b0[1] : w1b0[3];

    const int rb_local = tid & ~31;      // wave's first row within the tile

    // ---- software pipeline: prefetch first tile ----
    int t = blockIdx.x;
    if (t < ntiles) async_stage_row(x, t * TPB + tid, max_voff, lds0);

    unsigned cbuf = 0;
    for (; t < ntiles; t += tstride, cbuf ^= 1u) {
        const bool more = (t + tstride) < ntiles;
        if (more)
            async_stage_row(x, (t + tstride) * TPB + tid, max_voff,
                            cbuf ? lds0 : lds1);
        if (more) asm volatile("s_wait_asynccnt 0x1" ::: "memory");
        else      asm volatile("s_wait_asynccnt 0x0" ::: "memory");

        // x of group-0 row (base+m) and group-1 row (base+16+m); own row is one
        // of the two depending on the lane half.
        const v2f xg0 = cbuf ? sbuf[1][rb_local + m]      : sbuf[0][rb_local + m];
        const v2f xg1 = cbuf ? sbuf[1][rb_local + 16 + m] : sbuf[0][rb_local + 16 + m];
        const float x0 = lo ? xg0.x : xg1.x;   // own row
        const float x1 = lo ? xg0.y : xg1.y;

        // ---- mono0 (scalar, own row): h0(q)=q*relu(x0*wd) since temp==0,q>=0
        float rb[4];
#pragma unroll
        for (int k = 0; k < 4; ++k) rb[k] = fmaxf(x0 * w0d[k], 0.0f);
        float M0[4];
#pragma unroll
        for (int j = 0; j < 4; ++j) {
            float mm = rb[0] * w0s1[0 * 4 + j];
            mm = fmaf(rb[1], w0s1[1 * 4 + j], mm);
            mm = fmaf(rb[2], w0s1[2 * 4 + j], mm);
            mm = fmaf(rb[3], w0s1[3 * 4 + j], mm);
            M0[j] = mm;
        }

        // ---- mono1 group setup: this lane supplies h0 components k=kb,kb+1
        // for column m of group 0 (rows base..base+15) and group 1 (+16).
        const float baseA0 = xg0.y * wdA, baseB0 = xg0.y * wdB;
        const float baseA1 = xg1.y * wdA, baseB1 = xg1.y * wdB;
        const float tmpA0 = fmaf(xg0.x, s0A, b0A), tmpB0 = fmaf(xg0.x, s0B, b0B);
        const float tmpA1 = fmaf(xg1.x, s0A, b0A), tmpB1 = fmaf(xg1.x, s0B, b0B);

        // ---- fused quadrature loop ----
        float acc0 = 0.0f;            // mono0, own row (all 32 lanes)
        float accA = 0.0f;            // mono1 rows base+m   (valid lanes 0-15)
        float accB = 0.0f;            // mono1 rows base+16+m (valid lanes 0-15)
#pragma unroll
        for (int i = 0; i < NQ; ++i) {
            const float q = (float)i * 0.05f;

            // mono0 scalar path
            float f0 = w0bl;
#pragma unroll
            for (int j = 0; j < 4; ++j) {
                const float h = fmaxf(fmaf(q, M0[j], w0b1[j]), 0.0f);
                f0 = fmaf(h, w0l[j], f0);
            }
            acc0 = fmaf(f0, f0, acc0);

            // mono1: h0^T built directly in B-layout, then D = W1^T@h0^T + bias
            v2f B0, B1;
            B0.x = fmaxf(fmaf(q, baseA0, tmpA0), 0.0f);
            B0.y = fmaxf(fmaf(q, baseB0, tmpB0), 0.0f);
            B1.x = fmaxf(fmaf(q, baseA1, tmpA1), 0.0f);
            B1.y = fmaxf(fmaf(q, baseB1, tmpB1), 0.0f);

            v8f D0 = __builtin_amdgcn_wmma_f32_16x16x4_f32(
                false, Av, false, B0, (short)0, Cb, false, false);
            v8f D1 = __builtin_amdgcn_wmma_f32_16x16x4_f32(
                false, Av, false, B1, (short)0, Cb, false, false);

            // D[j] lane n = h1pre[row n, j] : relu + wlast dot, all in-lane.
            float fA = w1bl, fB = w1bl;
#pragma unroll
            for (int j = 0; j < 4; ++j) {
                fA = fmaf(fmaxf(D0[j], 0.0f), w1l[j], fA);
                fB = fmaf(fmaxf(D1[j], 0.0f), w1l[j], fB);
            }
            accA = fmaf(fA, fA, accA);
            accB = fmaf(fB, fB, accB);
        }

        // ---- generic 2-4-4-1 MLP on own (x0, x1)
        float gh[4];
#pragma unroll
        for (int j = 0; j < 4; ++j)
            gh[j] = fmaxf(fmaf(x1, gw0[4 + j], fmaf(x0, gw0[j], gb0[j])), 0.0f);
        float gout = gblv;
#pragma unroll
        for (int j = 0; j < 4; ++j) {
            float h = gb1[j];
            h = fmaf(gh[0], gw1[0 * 4 + j], h);
            h = fmaf(gh[1], gw1[1 * 4 + j], h);
            h = fmaf(gh[2], gw1[2 * 4 + j], h);
            h = fmaf(gh[3], gw1[3 * 4 + j], h);
            h = fmaxf(h, 0.0f);
            gout = fmaf(h, gwl[j], gout);
        }

        // Bring group-1 accumulators to the upper lane half (own-row layout).
        const float accBsw = __shfl_xor(accB, 16, 32);
        const float acc1 = lo ? accA : accBsw;

        const float wq = 0.05f;  // quadrature weight = 1/NQ
        v2f y;
        y.x = fmaf(wq * acc0, x0, g0bv);
        y.y = fmaf(wq * acc1, x1, gout);

        const int row = t * TPB + tid;
        if (row < n_rows)
            __builtin_nontemporal_store(y, (v2f*)&out[row]);
    }
}

extern "C" void kernel_launch(void* const* d_in, const int* in_sizes, int n_in,
                              void* d_out, int out_size, void* d_ws, size_t ws_size,
                              hipStream_t stream) {
    (void)n_in; (void)out_size; (void)d_ws; (void)ws_size;
    const int n_rows = in_sizes[0] / 2;  // x is (N, 2)
    const int ntiles = (n_rows + TPB - 1) / TPB;
    int grid = ntiles < 2048 ? ntiles : 2048;  // >=2 tiles/block -> live pipeline
    if (grid < 1) grid = 1;
    fused_mono_kernel<<<grid, TPB, 0, stream>>>(
        (const float2*)d_in[0],
        (const float*)d_in[1],  (const float*)d_in[2],  (const float*)d_in[3],
        (const float*)d_in[4],  (const float*)d_in[5],
        (const float*)d_in[6],  (const float*)d_in[7],  (const float*)d_in[8],
        (const float*)d_in[9],  (const float*)d_in[10], (const float*)d_in[11],
        (const float*)d_in[12],
        (const float*)d_in[13],
        (const float*)d_in[14], (const float*)d_in[15],
        (const float*)d_in[16], (const float*)d_in[17],
        (const float*)d_in[18], (const float*)d_in[19],
        (float2*)d_out, n_rows);
}